// KernalAnsatz_65481071396342
// MI455X (gfx1250) — compile-verified
//
#include <hip/hip_runtime.h>

typedef __attribute__((ext_vector_type(2))) float v2f;
typedef __attribute__((ext_vector_type(8))) float v8f;

#define BX 8192
#define BY 8192
#define FDIM 64
#define TILES_N (BY / 16)   // 512
#define TILES_M (BX / 16)   // 512
#define WAVES_PER_BLOCK 8

// One wave32 computes one 16x16 output tile:
//   cross = X[16x64] * Y[16x64]^T  via 16 chained V_WMMA_F32_16X16X4_F32
//   dist  = ||x||^2 + ||y||^2 - 2*cross   (norms computed from same fragments)
//   out   = exp(-gamma * max(dist, 0))
__global__ __launch_bounds__(256) void rbf_wmma_f32_kernel(
    const float* __restrict__ x, const float* __restrict__ y,
    const float* __restrict__ gamma, float* __restrict__ out) {
  const int lane = threadIdx.x & 31;
  const int wave = threadIdx.x >> 5;
  const int tile = blockIdx.x * WAVES_PER_BLOCK + wave;
  const int tm = tile / TILES_N;
  const int tn = tile - tm * TILES_N;
  const int row0 = tm * 16;
  const int col0 = tn * 16;

  const int lo = lane & 15;   // M index (A) / N index (B)
  const int hi = lane >> 4;   // selects K pair {0,1} vs {2,3} within 4-chunk

  // Per-lane source rows (row-major, 64 floats each).
  const float* xrow = x + (size_t)(row0 + lo) * FDIM + 2 * hi;
  const float* yrow = y + (size_t)(col0 + lo) * FDIM + 2 * hi;

  v8f acc = {};           // 16x16 f32 C/D: VGPR i holds row M = i + 8*hi, N = lo
  float xs = 0.0f;        // partial ||x_row(lo)||^2 over this lane's K-elements
  float ys = 0.0f;        // partial ||y_row(lo)||^2

#pragma unroll
  for (int kk = 0; kk < FDIM; kk += 4) {
    v2f a = *(const v2f*)(xrow + kk);   // A 16x4 f32 fragment (2 VGPRs)
    v2f b = *(const v2f*)(yrow + kk);   // B 4x16 f32 fragment (2 VGPRs)
    // (neg_a, A, neg_b, B, c_mod, C, reuse_a, reuse_b)
    acc = __builtin_amdgcn_wmma_f32_16x16x4_f32(
        false, a, false, b, (short)0, acc, false, false);
    xs = __builtin_fmaf(a.x, a.x, xs);
    xs = __builtin_fmaf(a.y, a.y, xs);
    ys = __builtin_fmaf(b.x, b.x, ys);
    ys = __builtin_fmaf(b.y, b.y, ys);
  }

  // Combine the two K-halves (lane L and L^16 hold complementary halves).
  xs += __shfl_xor(xs, 16, 32);   // full ||x[row0 + lo]||^2, both half-waves
  ys += __shfl_xor(ys, 16, 32);   // full ||y[col0 + lo]||^2 == y_sq[N] this lane needs

  const float g = gamma[0];       // uniform scalar (s_load)

  float* orow = out + (size_t)(row0 + 8 * hi) * BY + col0 + lo;
#pragma unroll
  for (int i = 0; i < 8; ++i) {
    // x_sq for accumulator row M = i + 8*hi lives in lane (i + 8*hi).
    float xsq_m = __shfl(xs, i + 8 * hi, 32);
    float d = __builtin_fmaf(-2.0f, acc[i], xsq_m + ys);
    d = fmaxf(d, 0.0f);
    orow[(size_t)i * BY] = __expf(-g * d);
  }
}

extern "C" void kernel_launch(void* const* d_in, const int* in_sizes, int n_in,
                              void* d_out, int out_size, void* d_ws, size_t ws_size,
                              hipStream_t stream) {
  const float* x = (const float*)d_in[0];
  const float* y = (const float*)d_in[1];
  const float* g = (const float*)d_in[2];
  float* out = (float*)d_out;

  const int total_tiles = TILES_M * TILES_N;          // 262144
  const int blocks = total_tiles / WAVES_PER_BLOCK;   // 32768
  rbf_wmma_f32_kernel<<<blocks, 32 * WAVES_PER_BLOCK, 0, stream>>>(x, y, g, out);
}